// GaussianKernel_584115552818
// MI455X (gfx1250) — compile-verified
//
#include <hip/hip_runtime.h>
#include <hip/hip_bf16.h>

typedef __attribute__((ext_vector_type(2))) float v2f;
typedef __attribute__((ext_vector_type(8))) float v8f;

#define M_PTS 4096
#define N_PTS 8192
#define D_FEAT 8

// -------- Kernel 0: squared L2 norms of every test / train point --------
__global__ __launch_bounds__(256) void kde_norms_kernel(
    const float* __restrict__ test, const float* __restrict__ train,
    float* __restrict__ nTest, float* __restrict__ nTrain) {
  int i = blockIdx.x * blockDim.x + threadIdx.x;
  if (i < M_PTS) {
    const float* p = test + i * D_FEAT;
    float s = 0.f;
#pragma unroll
    for (int k = 0; k < D_FEAT; ++k) s += p[k] * p[k];
    nTest[i] = s;
  }
  if (i < N_PTS) {
    const float* p = train + i * D_FEAT;
    float s = 0.f;
#pragma unroll
    for (int k = 0; k < D_FEAT; ++k) s += p[k] * p[k];
    nTrain[i] = s;
  }
}

// -------- Kernel 1: KDE via fp32 WMMA GEMM + exp epilogue --------
// Each block owns a 16-row test tile. 4 waves, each sweeps N_PTS/4 train
// columns in 16-wide tiles. Dot products via V_WMMA_F32_16X16X4_F32 (x2 for K=8).
__global__ __launch_bounds__(128) void kde_wmma_kernel(
    const float* __restrict__ test, const float* __restrict__ train,
    const float* __restrict__ nTest, const float* __restrict__ nTrain,
    float* __restrict__ out) {
  __shared__ float red[4 * 16];

  const int lane  = threadIdx.x & 31;
  const int wave  = threadIdx.x >> 5;
  const int half  = lane >> 4;     // 0: lanes 0-15, 1: lanes 16-31
  const int l16   = lane & 15;
  const int khalf = half * 2;      // K offset per ISA 32-bit A layout
  const int mBase = blockIdx.x * 16;

  // A tile (16x8 test rows), layout: lane%16 = M, VGPR pair = contiguous K pair
  const int m = mBase + l16;
  v2f a0 = *(const v2f*)(test + m * D_FEAT + khalf);       // K = khalf, khalf+1
  v2f a1 = *(const v2f*)(test + m * D_FEAT + khalf + 4);   // K = khalf+4, khalf+5

  // Test-point norms for the 8 C-rows this lane will hold (M = r + 8*half)
  float nA[8];
#pragma unroll
  for (int r = 0; r < 8; ++r) nA[r] = nTest[mBase + r + 8 * half];

  v8f acc = {};  // per-lane density partial sums, 8 M-rows x this lane's N set

  const int nBegin = wave * (N_PTS / 4);
  const int nEnd   = nBegin + (N_PTS / 4);
  for (int n0 = nBegin; n0 < nEnd; n0 += 16) {
    const int n = n0 + l16;
    // B tile (8xK x 16 train cols), symmetric layout: lane%16 = N
    v2f b0 = *(const v2f*)(train + n * D_FEAT + khalf);
    v2f b1 = *(const v2f*)(train + n * D_FEAT + khalf + 4);
    float nB = nTrain[n];

    v8f c = {};
    // D = A x B + C : two K=4 fp32 WMMAs accumulate the full K=8 dot product
    c = __builtin_amdgcn_wmma_f32_16x16x4_f32(false, a0, false, b0,
                                              (short)0, c, false, false);
    c = __builtin_amdgcn_wmma_f32_16x16x4_f32(false, a1, false, b1,
                                              (short)0, c, false, false);

    // dens = exp(-(|x|^2 + |y|^2 - 2 x.y) / (2*BW^2)); 1/(2*0.05^2) = 200
#pragma unroll
    for (int r = 0; r < 8; ++r) {
      float e = (2.0f * c[r] - nA[r] - nB) * 200.0f;
      acc[r] += __expf(e);
    }
  }

  // Reduce over N: butterfly within each 16-lane half (C layout: N = lane%16)
#pragma unroll
  for (int r = 0; r < 8; ++r) {
    float v = acc[r];
    v += __shfl_xor(v, 1, 32);
    v += __shfl_xor(v, 2, 32);
    v += __shfl_xor(v, 4, 32);
    v += __shfl_xor(v, 8, 32);
    acc[r] = v;
  }
  if (l16 == 0) {
#pragma unroll
    for (int r = 0; r < 8; ++r) red[wave * 16 + half * 8 + r] = acc[r];
  }
  __syncthreads();

  if (threadIdx.x < 16) {
    float s = red[threadIdx.x] + red[16 + threadIdx.x] +
              red[32 + threadIdx.x] + red[48 + threadIdx.x];
    // coef = 1/sqrt(2*pi*BW^2) = 7.9788456; mean over N train points
    out[mBase + threadIdx.x] = s * (7.978845608028654f / (float)N_PTS);
  }
}

extern "C" void kernel_launch(void* const* d_in, const int* in_sizes, int n_in,
                              void* d_out, int out_size, void* d_ws, size_t ws_size,
                              hipStream_t stream) {
  const float* test  = (const float*)d_in[0];   // [4096, 8]
  const float* train = (const float*)d_in[1];   // [8192, 8]
  float* out = (float*)d_out;                   // [4096]

  float* nTest  = (float*)d_ws;                 // [4096]
  float* nTrain = nTest + M_PTS;                // [8192]

  kde_norms_kernel<<<(N_PTS + 255) / 256, 256, 0, stream>>>(test, train, nTest, nTrain);
  kde_wmma_kernel<<<M_PTS / 16, 128, 0, stream>>>(test, train, nTest, nTrain, out);
}